// Fold2D_63909113364644
// MI455X (gfx1250) — compile-verified
//
#include <hip/hip_runtime.h>

// ---------------------------------------------------------------------------
// Fold2D, x: (8, 512, 512, 32) f32, L = 16, channels_last.
//
// y = x everywhere except a thin band:
//   h in [16,32):             y = x[h,w] + x[31-h,w]   (+ corner terms)
//   w in [16,32), h outside:  y = x[h,w] + x[h,31-w]
//
// Strategy (MI455X, 23.3 TB/s HBM):
//   Kernel 1: full-tensor NT b128 streaming copy  (bandwidth floor ~23 us)
//   Kernel 2: band fix-up as the *actual* flip-matrix matmul, using
//             V_WMMA_F32_16X16X4_F32 (D = J(16x16) @ X + C), 4 K-chunks.
// ---------------------------------------------------------------------------

typedef __attribute__((ext_vector_type(4))) float f4;
typedef __attribute__((ext_vector_type(2))) float v2f;
typedef __attribute__((ext_vector_type(8))) float v8f;

#define NB 8
#define NH 512
#define NW 512
#define NC 32
#define FL 16

__device__ __forceinline__ unsigned eidx(int b, int h, int w, int c) {
    return (((unsigned)b * NH + (unsigned)h) * NW + (unsigned)w) * NC + (unsigned)c;
}

// ---- Kernel 1: bulk streaming copy (b128, non-temporal) --------------------
__global__ void __launch_bounds__(256) fold2d_copy(const f4* __restrict__ in,
                                                   f4* __restrict__ out, int n4) {
    int i = blockIdx.x * blockDim.x + threadIdx.x;
    if (i < n4) {
        f4 v = __builtin_nontemporal_load(&in[i]);
        __builtin_nontemporal_store(v, &out[i]);
    }
}

// ---- Kernel 2: band fix-up via V_WMMA_F32_16X16X4_F32 ----------------------
// Block = 1024 threads = 32 waves; wave w handles channel c = w of one tile,
// so the 128B lines (32 channels of a pixel) are reused across waves in-WGP.
// Grid: first NB*32 blocks = row band (h in [16,32), 32 w-tiles),
//       next  NB*31 blocks = col band (w in [16,32), 31 h-tiles outside [16,32)).
__global__ void __launch_bounds__(1024) fold2d_band(const float* __restrict__ x,
                                                    float* __restrict__ y) {
    const int lane = threadIdx.x & 31;
    const int cch  = threadIdx.x >> 5;   // wave id == channel 0..31
    const int half = lane >> 4;          // 0: lanes 0-15, 1: lanes 16-31
    const int l16  = lane & 15;
    const int n    = l16;                // N index (column of the 16x16 tile)
    const int kk0  = half * 2;           // A/B K slot in vgpr .x  (ISA 7.12.2)
    const int kk1  = half * 2 + 1;       // A/B K slot in vgpr .y

    v8f acc;
    int blk = blockIdx.x;

    if (blk < NB * 32) {
        // ---- row band: D(16x16) = J @ Xtop(tile) + Xbot(tile) ----
        int b  = blk >> 5;
        int t  = blk & 31;
        int w0 = t * 16;
        bool corner = (t == 1);          // w-tile [16,32): fold W-reversal into loads

        // C init = Xbot (+ column-reversed Xbot for corner). C/D layout: vgpr r,
        // lanes 0-15 -> M=r, lanes 16-31 -> M=8+r.
#pragma unroll
        for (int r = 0; r < 8; ++r) {
            int m = r + half * 8;
            float v = x[eidx(b, FL + m, w0 + n, cch)];
            if (corner) v += x[eidx(b, FL + m, 15 - n, cch)];
            acc[r] = v;
        }
#pragma unroll
        for (int k = 0; k < 4; ++k) {
            // A = J chunk k (analytic): A[m,kk] = 1 iff 4k+kk == 15-m
            v2f a, bm;
            a.x = (4 * k + kk0 == 15 - l16) ? 1.0f : 0.0f;
            a.y = (4 * k + kk1 == 15 - l16) ? 1.0f : 0.0f;
            // B = Xtop chunk k: B[kk,n] = x[4k+kk, w0+n] (+ reversed for corner)
            int hb0 = 4 * k + kk0, hb1 = 4 * k + kk1;
            float b0 = x[eidx(b, hb0, w0 + n, cch)];
            float b1 = x[eidx(b, hb1, w0 + n, cch)];
            if (corner) {
                b0 += x[eidx(b, hb0, 15 - n, cch)];
                b1 += x[eidx(b, hb1, 15 - n, cch)];
            }
            bm.x = b0; bm.y = b1;
            acc = __builtin_amdgcn_wmma_f32_16x16x4_f32(
                false, a, false, bm, (short)0, acc, false, false);
        }
#pragma unroll
        for (int r = 0; r < 8; ++r) {
            int m = r + half * 8;
            y[eidx(b, FL + m, w0 + n, cch)] = acc[r];
        }
    } else {
        // ---- col band: D(16x16) = Xleft @ J + Xmid, h-tiles outside [16,32) ----
        int blk2 = blk - NB * 32;
        int b  = blk2 / 31;
        int ti = blk2 % 31;
        int h0 = (ti == 0) ? 0 : (FL + ti * 16);   // 0, 32, 48, ..., 496

        // C init = Xmid = x[h0+m, 16+n]
#pragma unroll
        for (int r = 0; r < 8; ++r) {
            int m = r + half * 8;
            acc[r] = x[eidx(b, h0 + m, FL + n, cch)];
        }
#pragma unroll
        for (int k = 0; k < 4; ++k) {
            v2f a, bm;
            // A = Xleft chunk k: A[m,kk] = x[h0+m, 4k+kk]
            a.x = x[eidx(b, h0 + l16, 4 * k + kk0, cch)];
            a.y = x[eidx(b, h0 + l16, 4 * k + kk1, cch)];
            // B = J chunk k (analytic): B[kk,n] = 1 iff 4k+kk == 15-n
            bm.x = (4 * k + kk0 == 15 - n) ? 1.0f : 0.0f;
            bm.y = (4 * k + kk1 == 15 - n) ? 1.0f : 0.0f;
            acc = __builtin_amdgcn_wmma_f32_16x16x4_f32(
                false, a, false, bm, (short)0, acc, false, false);
        }
#pragma unroll
        for (int r = 0; r < 8; ++r) {
            int m = r + half * 8;
            y[eidx(b, h0 + m, FL + n, cch)] = acc[r];
        }
    }
}

extern "C" void kernel_launch(void* const* d_in, const int* in_sizes, int n_in,
                              void* d_out, int out_size, void* d_ws, size_t ws_size,
                              hipStream_t stream) {
    const float* x = (const float*)d_in[0];
    float* y = (float*)d_out;

    // Kernel 1: copy everything (band pixels get overwritten by kernel 2,
    // which is ordered after it on the same stream).
    int n4 = in_sizes[0] / 4;                       // 16,777,216 float4 units
    int blocks = (n4 + 255) / 256;                  // 65,536 blocks
    fold2d_copy<<<blocks, 256, 0, stream>>>((const f4*)x, (f4*)y, n4);

    // Kernel 2: band fix-up. NB*32 row-band tiles + NB*31 col-band tiles,
    // 32 waves per block (one per channel).
    fold2d_band<<<NB * 32 + NB * 31, 1024, 0, stream>>>(x, y);
}